// GraphSAGE_82119774700011
// MI455X (gfx1250) — compile-verified
//
#include <hip/hip_runtime.h>
#include <math.h>

#define N_NODES 100000
#define N_EDGES 1600000
#define BB 8
#define GNN 64
#define N_LAYERS 5
#define OUT_CH 17
#define OUT_TOT 19

typedef __attribute__((ext_vector_type(2))) float v2f;
typedef __attribute__((ext_vector_type(8))) float v8f;

__device__ __forceinline__ float silu_f(float v) { return v / (1.0f + __expf(-v)); }

// ---------------------------------------------------------------------------
// CDNA5 async global->LDS DMA (ASYNCcnt-tracked).  VDST operand is the LDS
// byte address (low 32 bits of the generic shared pointer == LDS offset per
// the aperture mapping); VADDR is the 64-bit global address.  16B aligned.
// ---------------------------------------------------------------------------
__device__ __forceinline__ void async_copy_b128(void* lds_dst, const void* gsrc)
{
    unsigned off = (unsigned)(size_t)lds_dst;
    asm volatile("global_load_async_to_lds_b128 %0, %1, off"
                 :: "v"(off), "v"(gsrc)
                 : "memory");
}
__device__ __forceinline__ void wait_async0()
{
    asm volatile("s_wait_asynccnt 0x0" ::: "memory");
}

// ---------------------------------------------------------------------------
// K1: cond sincos embed -> 3-layer MLP -> latent (8x256); also precompute the
// per-layer FiLM tables gb[l] = latent @ fW[l] + fb[l]  (5 x 8 x 128).
// Tiny problem: one 256-thread block, plain VALU.
// ---------------------------------------------------------------------------
__global__ __launch_bounds__(256) void cond_mlp_kernel(
    const float* __restrict__ cond,
    const float* __restrict__ cW1, const float* __restrict__ cb1,
    const float* __restrict__ cW2, const float* __restrict__ cb2,
    const float* __restrict__ cW3, const float* __restrict__ cb3,
    const float* __restrict__ fW,  const float* __restrict__ fb,
    float* __restrict__ ws_lat, float* __restrict__ ws_gb,
    float* __restrict__ out_lat)
{
    __shared__ float s_emb[BB * 256];
    __shared__ float s_h1[BB * 128];
    __shared__ float s_h2[BB * 64];
    __shared__ float s_lat[BB * 256];
    const int t = threadIdx.x;
    const float LOG1E4 = 9.210340371976184f; // ln(10000)

    // sincos embed: dim=256, ndim=4 -> half=32
    for (int i = t; i < BB * 256; i += 256) {
        int b = i >> 8, f = i & 255;
        int c = f >> 6, r = f & 63, j = r & 31, isc = r >> 5;
        float om  = __expf(-((float)j / 32.0f) * LOG1E4);
        float ang = cond[b * 4 + c] * om;
        s_emb[i] = isc ? __cosf(ang) : __sinf(ang);
    }
    __syncthreads();
    for (int i = t; i < BB * 128; i += 256) {
        int b = i >> 7, o = i & 127;
        float acc = cb1[o];
        for (int k = 0; k < 256; ++k) acc += s_emb[b * 256 + k] * cW1[k * 128 + o];
        s_h1[i] = silu_f(acc);
    }
    __syncthreads();
    for (int i = t; i < BB * 64; i += 256) {
        int b = i >> 6, o = i & 63;
        float acc = cb2[o];
        for (int k = 0; k < 128; ++k) acc += s_h1[b * 128 + k] * cW2[k * 64 + o];
        s_h2[i] = silu_f(acc);
    }
    __syncthreads();
    for (int i = t; i < BB * 256; i += 256) {
        int b = i >> 8, o = i & 255;
        float acc = cb3[o];
        for (int k = 0; k < 64; ++k) acc += s_h2[b * 64 + k] * cW3[k * 256 + o];
        float v = silu_f(acc);
        s_lat[i] = v; ws_lat[i] = v; out_lat[i] = v;
    }
    __syncthreads();
    // gb[l] = latent @ fW[l] + fb[l]   (gather through batch_index is linear,
    // so this (8,256)@(256,128) replaces the reference's (100000,256) GEMM)
    for (int l = 0; l < N_LAYERS; ++l)
        for (int i = t; i < BB * 128; i += 256) {
            int b = i >> 7, o = i & 127;
            float acc = fb[l * 128 + o];
            for (int k = 0; k < 256; ++k)
                acc += s_lat[b * 256 + k] * fW[(l * 256 + k) * 128 + o];
            ws_gb[l * BB * 128 + i] = acc;
        }
}

// ---------------------------------------------------------------------------
// Degree kernels
// ---------------------------------------------------------------------------
__global__ void zero_kernel(float* __restrict__ p, int n)
{
    int i = blockIdx.x * blockDim.x + threadIdx.x;
    if (i < n) p[i] = 0.0f;
}

__global__ void deg_kernel(const int* __restrict__ dst, float* __restrict__ deg)
{
    for (int e = blockIdx.x * blockDim.x + threadIdx.x; e < N_EDGES;
         e += gridDim.x * blockDim.x)
        atomicAdd(&deg[dst[e]], 1.0f);
}

__global__ void invdeg_kernel(float* __restrict__ deg)
{
    int i = blockIdx.x * blockDim.x + threadIdx.x;
    if (i < N_NODES) {
        float d = deg[i];
        deg[i] = (d > 0.0f) ? (1.0f / d) : 0.0f;
    }
}

// ---------------------------------------------------------------------------
// K3: node embed:  x = sincos_embed(coords, 64) @ eW + eb   via fp32 WMMA.
// 128 threads = 4 waves; each wave owns a 16-node tile; K=64 chained in
// 16 steps of v_wmma_f32_16x16x4_f32; N=64 as 4 tiles of 16.
// eW is staged to LDS with async global->LDS DMA.
// ---------------------------------------------------------------------------
__global__ __launch_bounds__(128) void node_embed_kernel(
    const float* __restrict__ coords, const float* __restrict__ eW,
    const float* __restrict__ eb, float* __restrict__ x)
{
    __shared__ __align__(16) float s_w[64 * 64];    // eW, row-major [k][n]
    __shared__ __align__(16) float s_e[4][16 * 64]; // per-wave 16x64 A tile
    const int t = threadIdx.x, w = t >> 5, lane = t & 31;

    // async DMA the 16KB weight tile: 128 threads x b128
    for (int i = t * 4; i < 4096; i += 128 * 4)
        async_copy_b128(&s_w[i], eW + i);

    const int base = blockIdx.x * 64 + w * 16;
    const float LOG1E4 = 9.210340371976184f;
    // sincos embed: dim=64, ndim=2 -> half=16 (VALU work overlaps the DMA)
    for (int q = 0; q < 32; ++q) {
        int idx = lane + q * 32;                 // 0..1023
        int m = idx >> 6, f = idx & 63;
        int node = base + m; if (node >= N_NODES) node = N_NODES - 1;
        int c = f >> 5, r = f & 31, j = r & 15, isc = r >> 4;
        float om  = __expf(-((float)j / 16.0f) * LOG1E4);
        float ang = coords[node * 2 + c] * om;
        s_e[w][idx] = isc ? __cosf(ang) : __sinf(ang);
    }
    wait_async0();
    __syncthreads();

    const int half = lane >> 4, l16 = lane & 15;
    for (int nt = 0; nt < 4; ++nt) {
        v8f acc = {};
        const int col = nt * 16 + l16;
        for (int k0 = 0; k0 < 64; k0 += 4) {
            int kb = k0 + half * 2;
            v2f a, b;
            a.x = s_e[w][l16 * 64 + kb];
            a.y = s_e[w][l16 * 64 + kb + 1];
            b.x = s_w[kb * 64 + col];
            b.y = s_w[(kb + 1) * 64 + col];
            acc = __builtin_amdgcn_wmma_f32_16x16x4_f32(
                false, a, false, b, (short)0, acc, false, false);
        }
        float bias = eb[col];
#pragma unroll
        for (int r = 0; r < 8; ++r) {
            int node = base + half * 8 + r;
            if (node < N_NODES) x[node * 64 + col] = acc[r] + bias;
        }
    }
}

// ---------------------------------------------------------------------------
// K4a: FiLM modulation + zero the aggregation buffer in the same pass.
// ---------------------------------------------------------------------------
__global__ void film_kernel(const float* __restrict__ x,
                            const int* __restrict__ bidx,
                            const float* __restrict__ gb,   // [8][128] for layer l
                            float* __restrict__ xm, float* __restrict__ agg)
{
    const int total = N_NODES * 64;
    for (int i = blockIdx.x * blockDim.x + threadIdx.x; i < total;
         i += gridDim.x * blockDim.x) {
        int n = i >> 6, f = i & 63;
        int b = bidx[n];
        float g  = gb[b * 128 + f];
        float be = gb[b * 128 + 64 + f];
        xm[i]  = x[i] * (1.0f + g) + be;
        agg[i] = 0.0f;
    }
}

// ---------------------------------------------------------------------------
// K4b: edge scatter: agg[dst] += xm[src].  16 threads/edge, float4 gathers
// (xm/agg are ~25.6MB each -> resident in the 192MB L2).
// ---------------------------------------------------------------------------
__global__ void scatter_kernel(const int* __restrict__ src,
                               const int* __restrict__ dst,
                               const float* __restrict__ xm,
                               float* __restrict__ agg)
{
    const long long total = (long long)N_EDGES * 16;
    for (long long i = blockIdx.x * (long long)blockDim.x + threadIdx.x; i < total;
         i += (long long)gridDim.x * blockDim.x) {
        int e  = (int)(i >> 4);
        int f4 = ((int)i & 15) << 2;
        int s = src[e], d = dst[e];
        float4 v = *(const float4*)(xm + (size_t)s * 64 + f4);
        float* ap = agg + (size_t)d * 64 + f4;
        atomicAdd(ap + 0, v.x);
        atomicAdd(ap + 1, v.y);
        atomicAdd(ap + 2, v.z);
        atomicAdd(ap + 3, v.w);
    }
}

// ---------------------------------------------------------------------------
// K4c: x = silu( (agg*inv_deg) @ lW + lb + xm @ rW )   via fp32 WMMA.
// 64 threads = 2 waves (48KB LDS).  All four LDS tiles (two 16KB weight
// tiles + per-wave agg/xm activation tiles) arrive via async global->LDS
// DMA; inv_deg is applied at A-fragment read time (one row per lane).
// ---------------------------------------------------------------------------
__global__ __launch_bounds__(64) void update_kernel(
    const float* __restrict__ agg, const float* __restrict__ xm,
    const float* __restrict__ invdeg,
    const float* __restrict__ lW, const float* __restrict__ lb,
    const float* __restrict__ rW, float* __restrict__ x)
{
    __shared__ __align__(16) float s_lw[4096];
    __shared__ __align__(16) float s_rw[4096];
    __shared__ __align__(16) float s_a[2][1024];
    __shared__ __align__(16) float s_m[2][1024];
    const int t = threadIdx.x, w = t >> 5, lane = t & 31;

    // async DMA both 16KB weight tiles (64 threads x b128)
    for (int i = t * 4; i < 4096; i += 64 * 4) {
        async_copy_b128(&s_lw[i], lW + i);
        async_copy_b128(&s_rw[i], rW + i);
    }

    // async DMA this wave's 16x64 agg / xm tiles (raw; inv_deg applied later)
    const int base = blockIdx.x * 32 + w * 16;
    for (int q = 0; q < 8; ++q) {
        int idx = (lane + q * 32) * 4;           // float index, 16B aligned
        int m = idx >> 6, f = idx & 63;
        int node = base + m; if (node >= N_NODES) node = N_NODES - 1;
        async_copy_b128(&s_a[w][idx], agg + (size_t)node * 64 + f);
        async_copy_b128(&s_m[w][idx], xm  + (size_t)node * 64 + f);
    }
    wait_async0();
    __syncthreads();

    const int half = lane >> 4, l16 = lane & 15;
    int arow = base + l16; if (arow >= N_NODES) arow = N_NODES - 1;
    const float id = invdeg[arow];               // one A-row per lane

    for (int nt = 0; nt < 4; ++nt) {
        v8f acc = {};
        const int col = nt * 16 + l16;
        for (int k0 = 0; k0 < 64; k0 += 4) {
            int kb = k0 + half * 2;
            v2f a1, b1, a2, b2;
            a1.x = s_a[w][l16 * 64 + kb] * id;
            a1.y = s_a[w][l16 * 64 + kb + 1] * id;
            b1.x = s_lw[kb * 64 + col];
            b1.y = s_lw[(kb + 1) * 64 + col];
            acc = __builtin_amdgcn_wmma_f32_16x16x4_f32(
                false, a1, false, b1, (short)0, acc, false, false);
            a2.x = s_m[w][l16 * 64 + kb];
            a2.y = s_m[w][l16 * 64 + kb + 1];
            b2.x = s_rw[kb * 64 + col];
            b2.y = s_rw[(kb + 1) * 64 + col];
            acc = __builtin_amdgcn_wmma_f32_16x16x4_f32(
                false, a2, false, b2, (short)0, acc, false, false);
        }
        float bias = lb[col];
#pragma unroll
        for (int r = 0; r < 8; ++r) {
            int node = base + half * 8 + r;
            if (node < N_NODES) x[node * 64 + col] = silu_f(acc[r] + bias);
        }
    }
}

// ---------------------------------------------------------------------------
// K5: decode: out = x @ dW + db;  y = out[:, :17];  coords += out[:, 17:19]
// ---------------------------------------------------------------------------
__global__ __launch_bounds__(256) void decode_kernel(
    const float* __restrict__ x, const float* __restrict__ dW,
    const float* __restrict__ db, const float* __restrict__ coords,
    float* __restrict__ out)
{
    __shared__ float s_w[64 * OUT_TOT];
    __shared__ float s_b[OUT_TOT];
    const int t = threadIdx.x;
    for (int i = t; i < 64 * OUT_TOT; i += 256) s_w[i] = dW[i];
    if (t < OUT_TOT) s_b[t] = db[t];
    __syncthreads();

    const long long total = (long long)N_NODES * OUT_TOT;
    for (long long i = blockIdx.x * (long long)blockDim.x + t; i < total;
         i += (long long)gridDim.x * blockDim.x) {
        int n  = (int)(i / OUT_TOT);
        int ch = (int)(i % OUT_TOT);
        float acc = s_b[ch];
        const float* xr = x + (size_t)n * 64;
        for (int k = 0; k < 64; ++k) acc += xr[k] * s_w[k * OUT_TOT + ch];
        if (ch < OUT_CH)
            out[(size_t)n * OUT_CH + ch] = acc;                       // y
        else
            out[1700000 + (size_t)n * 2 + (ch - OUT_CH)] =            // coords
                coords[n * 2 + (ch - OUT_CH)] + acc;
    }
}

// ---------------------------------------------------------------------------
extern "C" void kernel_launch(void* const* d_in, const int* in_sizes, int n_in,
                              void* d_out, int out_size, void* d_ws, size_t ws_size,
                              hipStream_t stream)
{
    (void)in_sizes; (void)n_in; (void)out_size; (void)ws_size;
    const float* cond   = (const float*)d_in[0];
    const float* coords = (const float*)d_in[1];
    const int*   edges  = (const int*)d_in[2];
    const int*   bidx   = (const int*)d_in[3];
    const float* cW1 = (const float*)d_in[4];
    const float* cb1 = (const float*)d_in[5];
    const float* cW2 = (const float*)d_in[6];
    const float* cb2 = (const float*)d_in[7];
    const float* cW3 = (const float*)d_in[8];
    const float* cb3 = (const float*)d_in[9];
    const float* eW  = (const float*)d_in[10];
    const float* eb  = (const float*)d_in[11];
    const float* fW  = (const float*)d_in[12];
    const float* fb  = (const float*)d_in[13];
    const float* lW  = (const float*)d_in[14];
    const float* lb  = (const float*)d_in[15];
    const float* rW  = (const float*)d_in[16];
    const float* dW  = (const float*)d_in[17];
    const float* db  = (const float*)d_in[18];
    float* out = (float*)d_out;
    float* ws  = (float*)d_ws;

    // Workspace layout (floats): ~77 MB total
    float* ws_lat = ws;                    // 2048
    float* ws_gb  = ws + 2048;             // 5*8*128 = 5120
    float* ws_deg = ws + 8192;             // 100000 (becomes inv_deg)
    float* ws_x   = ws + 108544;           // 100000*64
    float* ws_xm  = ws_x  + 6400000;       // 100000*64
    float* ws_agg = ws_xm + 6400000;       // 100000*64

    const int* e_src = edges;
    const int* e_dst = edges + N_EDGES;

    cond_mlp_kernel<<<1, 256, 0, stream>>>(cond, cW1, cb1, cW2, cb2, cW3, cb3,
                                           fW, fb, ws_lat, ws_gb, out + 1900000);

    zero_kernel<<<(N_NODES + 255) / 256, 256, 0, stream>>>(ws_deg, N_NODES);
    deg_kernel<<<2048, 256, 0, stream>>>(e_dst, ws_deg);
    invdeg_kernel<<<(N_NODES + 255) / 256, 256, 0, stream>>>(ws_deg);

    node_embed_kernel<<<(N_NODES + 63) / 64, 128, 0, stream>>>(coords, eW, eb, ws_x);

    for (int l = 0; l < N_LAYERS; ++l) {
        film_kernel<<<4096, 256, 0, stream>>>(ws_x, bidx, ws_gb + l * BB * 128,
                                              ws_xm, ws_agg);
        scatter_kernel<<<8192, 256, 0, stream>>>(e_src, e_dst, ws_xm, ws_agg);
        update_kernel<<<(N_NODES + 31) / 32, 64, 0, stream>>>(
            ws_agg, ws_xm, ws_deg, lW + l * 4096, lb + l * 64, rW + l * 4096, ws_x);
    }

    decode_kernel<<<4096, 256, 0, stream>>>(ws_x, dW, db, coords, out);
}